// LayeredNANDGraph_79654463472042
// MI455X (gfx1250) — compile-verified
//
#include <hip/hip_runtime.h>
#include <hip/hip_bf16.h>

// ---------------------------------------------------------------------------
// LayeredNANDGraph, bit-packed + fully LDS-fused for MI455X (gfx1250).
//   W = 65536 bits -> 2048 u32 words per neuron row.
//   Each workgroup owns an 8-word (256-bit) column slice and runs all 33
//   layers in LDS (two 4096-row x stride-9 buffers = 288 KB of the WGP's
//   320 KB). Input slice is staged by TENSOR_LOAD_TO_LDS (with LDS padding
//   producing the stride-9 anti-bank-conflict layout for free); the result
//   slice is drained by TENSOR_STORE_FROM_LDS. A final kernel unpacks the
//   packed bits to f32 with fully coalesced 128B-per-thread stores.
// ---------------------------------------------------------------------------

#define NUM_IN    1024
#define NUM_OUT   1024
#define NROWS     4096
#define WORDS     2048        // 65536 bits / 32
#define WPC       8           // words per chunk (256 bits per workgroup)
#define STR       9           // padded LDS row stride in words (bank scramble)
#define NMID      31          // NUM_LAYERS - 1 middle layers
#define NTHREADS  1024
#define NBLOCKS   (WORDS / WPC)   // 256 workgroups
#define SMEM_BYTES (2 * NROWS * STR * 4)   // 294912 <= 320 KB WGP LDS

typedef unsigned int u32x4 __attribute__((ext_vector_type(4)));
typedef unsigned int u32x8 __attribute__((ext_vector_type(8)));

// Pack 4 bool bytes (0/1) in a u32 into 4 bits. Normalize then multiply-trick.
__device__ __forceinline__ unsigned pack4(unsigned v) {
    v &= 0x01010101u;                    // keeps only bit 0 of each byte (expects 0/1)
    return (v * 0x01020408u) >> 24;      // b0|b1<<1|b2<<2|b3<<3
}

// ---- Kernel 1: bool bytes [1024][65536] -> packed u32 [1024][2048] ---------
__global__ void pack_bits(const unsigned char* __restrict__ in,
                          unsigned* __restrict__ outp) {
    const unsigned t = blockIdx.x * blockDim.x + threadIdx.x;  // row*2048+word
    if (t >= NUM_IN * WORDS) return;
    const uint4* p = (const uint4*)(in + (size_t)t * 32);      // 32 bytes
    const uint4 a = p[0];
    const uint4 b = p[1];
    const unsigned r = pack4(a.x)        | (pack4(a.y) << 4)  |
                       (pack4(a.z) << 8) | (pack4(a.w) << 12) |
                       (pack4(b.x) << 16)| (pack4(b.y) << 20) |
                       (pack4(b.z) << 24)| (pack4(b.w) << 28);
    outp[t] = r;
}

// ---- TDM descriptor helper: 2D tile 8 words x 1024 rows, row stride 2048 ---
// pad_dwords != 0 adds LDS padding (loads only): 1 dword after every 8.
__device__ __forceinline__ void make_tdm_desc(u32x4& g0, u32x8& g1,
                                              unsigned lds_addr,
                                              unsigned long long gaddr,
                                              bool pad) {
    g0.x = 1u;                                    // count=1 valid descriptor
    g0.y = lds_addr;                              // LDS byte address
    g0.z = (unsigned)gaddr;                       // global_addr[31:0]
    g0.w = ((unsigned)(gaddr >> 32) & 0x01FFFFFFu)// global_addr[56:32]
         | (2u << 30);                            // type=2 ("image")

    g1.s0 = (2u << 16)                            // data_size = 4 bytes
          | (pad ? ((1u << 20) | (2u << 22)) : 0u); // pad_en, every 8dw, +1dw
    g1.s1 = (2048u & 0xFFFFu) << 16;              // tensor_dim0 = 2048 (lo16)
    g1.s2 = (1024u & 0xFFFFu) << 16;              // tensor_dim1 = 1024 (lo16)
    g1.s3 = (8u    & 0xFFFFu) << 16;              // tile_dim0   = 8
    g1.s4 = 1024u;                                // tile_dim1=1024, tile_dim2=0
    g1.s5 = 2048u;                                // tensor_dim0_stride[31:0]
    g1.s6 = 0u;
    g1.s7 = 0u;
}

// ---- Kernel 2: all 33 NAND layers fused in LDS -----------------------------
extern __shared__ unsigned smem[];   // [2][NROWS*STR]

__global__ void
__launch_bounds__(NTHREADS)
nand_layers(const unsigned* __restrict__ packed,          // [1024][2048] in
            unsigned* __restrict__ packed_out,            // [1024][2048] out
            const int*           __restrict__ idx_first,  // [4096][2]
            const unsigned char* __restrict__ mask_first, // [4096]
            const int*           __restrict__ idx_mid,    // [31][4096][2]
            const unsigned char* __restrict__ mask_mid,   // [31][4096]
            const int*           __restrict__ idx_last,   // [1024][2]
            const unsigned char* __restrict__ mask_last)  // [1024]
{
    unsigned* b0 = smem;
    unsigned* b1 = smem + NROWS * STR;
    const int tid = threadIdx.x;
    const unsigned w0 = blockIdx.x * WPC;   // first word of this column slice

    // ---- Stage 0: TDM DMA of input slice (1024 rows x 8 words) into b0 ----
    if (tid < 32) {   // wave 0 issues the single TDM op
        u32x4 g0; u32x8 g1;
        u32x4 gz; gz.x = 0u; gz.y = 0u; gz.z = 0u; gz.w = 0u;
        make_tdm_desc(g0, g1,
                      (unsigned)(unsigned long long)b0,        // LDS offset
                      (unsigned long long)(const void*)(packed + w0),
                      /*pad=*/true);
        asm volatile("tensor_load_to_lds %0, %1, %2, %3"
                     :
                     : "s"(g0), "s"(g1), "s"(gz), "s"(gz)
                     : "memory");
        __builtin_amdgcn_s_wait_tensorcnt(0);
    }
    __syncthreads();

    // ---- generic NAND layer: dst[n] = (src[i0] & src[i1]) ^ (mask? ~0:0) --
    auto layer = [&](const unsigned* __restrict__ src,
                     unsigned* __restrict__ dst,
                     const int* __restrict__ idx,
                     const unsigned char* __restrict__ mask,
                     int nout, int dstStride) {
        for (int n = tid; n < nout; n += NTHREADS) {
            const int2 p = ((const int2*)idx)[n];          // coalesced b64
            const unsigned inv = mask[n] ? 0xFFFFFFFFu : 0u;
            const unsigned* ra = src + (size_t)p.x * STR;
            const unsigned* rb = src + (size_t)p.y * STR;
            unsigned* d = dst + (size_t)n * dstStride;
#pragma unroll
            for (int w = 0; w < WPC; ++w)
                d[w] = (ra[w] & rb[w]) ^ inv;
        }
        __syncthreads();
    };

    // first layer: 1024 -> 4096
    layer(b0, b1, idx_first, mask_first, NROWS, STR);

    // 31 middle layers, ping-pong
    unsigned* src = b1;
    unsigned* dst = b0;
    for (int l = 0; l < NMID; ++l) {
        if (tid == 0 && l + 1 < NMID)   // warm L2 for next layer's indices
            __builtin_prefetch(idx_mid + (size_t)(l + 1) * NROWS * 2, 0, 1);
        layer(src, dst,
              idx_mid + (size_t)l * NROWS * 2,
              mask_mid + (size_t)l * NROWS, NROWS, STR);
        unsigned* t = src; src = dst; dst = t;
    }

    // last layer: 4096 -> 1024, written COMPACT (stride 8) so the TDM store
    // (which has no de-padding) can drain it as a dense 8x1024 tile.
    layer(src, dst, idx_last, mask_last, NUM_OUT, WPC);

    // ---- Stage N: TDM DMA of result slice (1024 rows x 8 words) to global -
    if (tid < 32) {
        u32x4 g0; u32x8 g1;
        u32x4 gz; gz.x = 0u; gz.y = 0u; gz.z = 0u; gz.w = 0u;
        make_tdm_desc(g0, g1,
                      (unsigned)(unsigned long long)dst,       // LDS offset
                      (unsigned long long)(void*)(packed_out + w0),
                      /*pad=*/false);
        asm volatile("tensor_store_from_lds %0, %1, %2, %3"
                     :
                     : "s"(g0), "s"(g1), "s"(gz), "s"(gz)
                     : "memory");
        __builtin_amdgcn_s_wait_tensorcnt(0);
    }
    // implicit S_WAIT_IDLE at S_ENDPGM covers remaining waves
}

// ---- Kernel 3: packed u32 [1024][2048] -> f32 [1024][65536] ----------------
// Each thread expands one u32 into 32 consecutive floats (128 B burst).
__global__ void unpack_f32(const unsigned* __restrict__ packed_out,
                           float* __restrict__ out) {
    const unsigned t = blockIdx.x * blockDim.x + threadIdx.x;  // row*2048+word
    if (t >= NUM_OUT * WORDS) return;
    const unsigned word = packed_out[t];
    float4* o = (float4*)(out + (size_t)t * 32);
#pragma unroll
    for (int q = 0; q < 8; ++q) {
        float4 v;
        v.x = (word >> (q * 4 + 0)) & 1u ? 1.0f : 0.0f;
        v.y = (word >> (q * 4 + 1)) & 1u ? 1.0f : 0.0f;
        v.z = (word >> (q * 4 + 2)) & 1u ? 1.0f : 0.0f;
        v.w = (word >> (q * 4 + 3)) & 1u ? 1.0f : 0.0f;
        o[q] = v;
    }
}

// ---------------------------------------------------------------------------
extern "C" void kernel_launch(void* const* d_in, const int* in_sizes, int n_in,
                              void* d_out, int out_size, void* d_ws, size_t ws_size,
                              hipStream_t stream) {
    const unsigned char* in_bits    = (const unsigned char*)d_in[0];
    const int*           idx_first  = (const int*)d_in[1];
    const unsigned char* mask_first = (const unsigned char*)d_in[2];
    const int*           idx_mid    = (const int*)d_in[3];
    const unsigned char* mask_mid   = (const unsigned char*)d_in[4];
    const int*           idx_last   = (const int*)d_in[5];
    const unsigned char* mask_last  = (const unsigned char*)d_in[6];
    float*    out        = (float*)d_out;
    unsigned* packed_in  = (unsigned*)d_ws;                      // 8 MiB
    unsigned* packed_out = packed_in + (size_t)NUM_IN * WORDS;   // 8 MiB

    // Kernel 1: pack bools -> bits
    const int total_words = NUM_IN * WORDS;
    pack_bits<<<(total_words + 255) / 256, 256, 0, stream>>>(in_bits, packed_in);

    // Kernel 2: fused 33-layer NAND network (288 KB dynamic LDS per WGP)
    hipFuncSetAttribute((const void*)nand_layers,
                        hipFuncAttributeMaxDynamicSharedMemorySize, SMEM_BYTES);
    nand_layers<<<NBLOCKS, NTHREADS, SMEM_BYTES, stream>>>(
        packed_in, packed_out, idx_first, mask_first, idx_mid, mask_mid,
        idx_last, mask_last);

    // Kernel 3: unpack bits -> f32 output (256 MB, pure bandwidth)
    const int total_out_words = NUM_OUT * WORDS;
    unpack_f32<<<(total_out_words + 255) / 256, 256, 0, stream>>>(packed_out, out);
}